// GNNModel_38766374814015
// MI455X (gfx1250) — compile-verified
//
#include <hip/hip_runtime.h>
#include <hip/hip_bf16.h>
#include <math.h>
#include <stdint.h>

typedef __attribute__((ext_vector_type(2))) float v2f;
typedef __attribute__((ext_vector_type(8))) float v8f;

#define EPS_LN 1e-5f
#define NS 12   // padded node-feature row stride (9 -> 12 floats, 48B, 16B-aligned)

__device__ __forceinline__ float leaky(float x, float s) { return x > 0.f ? x : s * x; }

// float atomic max via signed/unsigned integer ordering trick
__device__ __forceinline__ void atomicMaxF(float* addr, float val) {
    if (val >= 0.f) atomicMax((int*)addr, __float_as_int(val));
    else            atomicMin((unsigned int*)addr, (unsigned int)__float_as_int(val));
}

// ---- optional gfx1250 async Global->LDS copy path (ASYNCcnt-tracked) ----
#if defined(__gfx1250__) && __has_builtin(__builtin_amdgcn_global_load_async_to_lds_b32) && \
    __has_builtin(__builtin_amdgcn_s_wait_asynccnt)
#define HAVE_ASYNC_LDS 1
typedef __attribute__((address_space(1))) int g_as_int;
typedef __attribute__((address_space(3))) int l_as_int;
#endif

__device__ __forceinline__ void g2l_b32(const float* g, float* lds) {
#ifdef HAVE_ASYNC_LDS
    __builtin_amdgcn_global_load_async_to_lds_b32(
        (g_as_int*)(unsigned long long)(uintptr_t)g,
        (l_as_int*)(unsigned)(uintptr_t)lds, 0, 0);
#else
    *lds = *g;
#endif
}
__device__ __forceinline__ void g2l_wait() {
#ifdef HAVE_ASYNC_LDS
    __builtin_amdgcn_s_wait_asynccnt(0);
#endif
}

// ---------------------------------------------------------------- utility
__global__ void k_fill(float* __restrict__ p, float v, int n) {
    int i = blockIdx.x * blockDim.x + threadIdx.x;
    if (i < n) p[i] = v;
}

// ---------------------------------------------------------------- LN on x -> h0  [N, NS] (pad zeros)
__global__ void k_ln9(const float* __restrict__ x, const float* __restrict__ g,
                      const float* __restrict__ b, float* __restrict__ o, int N) {
    int t = blockIdx.x * blockDim.x + threadIdx.x;
    if (t >= N) return;
    float f[12], m = 0.f, v = 0.f;
#pragma unroll
    for (int k = 0; k < 9; ++k) { f[k] = x[t * 9 + k]; m += f[k]; }
    m *= (1.f / 9.f);
#pragma unroll
    for (int k = 0; k < 9; ++k) { float d = f[k] - m; v += d * d; }
    v *= (1.f / 9.f);
    float inv = rsqrtf(v + EPS_LN);
#pragma unroll
    for (int k = 0; k < 9; ++k) f[k] = (f[k] - m) * inv * g[k] + b[k];
    f[9] = f[10] = f[11] = 0.f;
    float4* op = (float4*)(o + (size_t)t * NS);
    op[0] = *(float4*)&f[0]; op[1] = *(float4*)&f[4]; op[2] = *(float4*)&f[8];
}

// ---------------------------------------------------------------- degree + loop-attr sum
__global__ void k_deg(const int* __restrict__ dst, const float* __restrict__ ea,
                      float* __restrict__ deg, float* __restrict__ loopsum, int E) {
    int i = blockIdx.x * blockDim.x + threadIdx.x;
    if (i >= E) return;
    int d = dst[i];
    atomicAdd(&deg[d], 1.f);
    atomicAdd(&loopsum[d], ea[i]);
}

__global__ void k_loopattr(float* __restrict__ loopsum, const float* __restrict__ deg, int N) {
    int t = blockIdx.x * blockDim.x + threadIdx.x;
    if (t >= N) return;
    loopsum[t] = loopsum[t] / fmaxf(deg[t], 1.f);
}

// ---------------------------------------------------------------- node projection xp = h @ W (9x9)
// WMMA f32 16x16x4: one wave = 16 rows; K=9 padded to 12 (3 chunks of 4).
__global__ void k_proj(const float* __restrict__ h, const float* __restrict__ W,
                       float* __restrict__ xp, int N) {
    __shared__ float Wp[12][16];     // B matrix, zero padded
    __shared__ float At[8][16][12];  // per-wave A tiles (rows 16B-aligned: 48B stride)
    int tid = threadIdx.x;
    int w = tid >> 5, lane = tid & 31, half = lane >> 4, l16 = lane & 15;
    int rowBase = blockIdx.x * 128 + w * 16;

    for (int i = tid; i < 12 * 16; i += 256) {
        int k = i >> 4, nn = i & 15;
        if (k < 9 && nn < 9) g2l_b32(&W[k * 9 + nn], &Wp[k][nn]);  // async G->LDS
        else                 Wp[k][nn] = 0.f;
    }
    if (half == 0) {
        int r = rowBase + l16;
        float4 r0 = {0,0,0,0}, r1 = {0,0,0,0}, r2 = {0,0,0,0};
        if (r < N) {
            const float4* hp = (const float4*)(h + (size_t)r * NS);
            r0 = hp[0]; r1 = hp[1]; r2 = hp[2];   // pads already zero
        }
        float4* at = (float4*)&At[w][l16][0];
        at[0] = r0; at[1] = r1; at[2] = r2;
    }
    g2l_wait();
    __syncthreads();

    v8f c = {};
#pragma unroll
    for (int ch = 0; ch < 3; ++ch) {
        int kb = ch * 4 + half * 2;   // A: lane half selects K-pair (ISA 16x4 f32 layout)
        v2f A; A.x = At[w][l16][kb]; A.y = At[w][l16][kb + 1];
        v2f B; B.x = Wp[kb][l16];    B.y = Wp[kb + 1][l16];
        c = __builtin_amdgcn_wmma_f32_16x16x4_f32(false, A, false, B, (short)0, c, false, false);
    }
#pragma unroll
    for (int j = 0; j < 8; ++j) {     // C: VGPR j -> rows j (lanes 0-15) / j+8 (lanes 16-31)
        int r = rowBase + j + half * 8;
        if (r < N && l16 < NS) xp[(size_t)r * NS + l16] = (l16 < 9) ? c[j] : 0.f;
    }
}

// ---------------------------------------------------------------- per-node attention scores
__global__ void k_scores(const float* __restrict__ xp, const float* __restrict__ as,
                         const float* __restrict__ ad, float* __restrict__ ss,
                         float* __restrict__ sd, int N) {
    int t = blockIdx.x * blockDim.x + threadIdx.x;
    if (t >= N) return;
    const float4* xr = (const float4*)(xp + (size_t)t * NS);
    float4 r0 = xr[0], r1 = xr[1], r2 = xr[2];
    float f[12];
    *(float4*)&f[0] = r0; *(float4*)&f[4] = r1; *(float4*)&f[8] = r2;
    float s0 = 0.f, s1 = 0.f;
#pragma unroll
    for (int k = 0; k < 9; ++k) { s0 += f[k] * as[k]; s1 += f[k] * ad[k]; }
    ss[t] = s0; sd[t] = s1;
}

// ---------------------------------------------------------------- edge pass 1: logits + segment max
__global__ void k_edge_a(const int* __restrict__ src, const int* __restrict__ dst,
                         const float* __restrict__ ea, const float* __restrict__ loop_attr,
                         const float* __restrict__ ss, const float* __restrict__ sd,
                         const float* __restrict__ We, const float* __restrict__ ae,
                         float* __restrict__ a_buf, float* __restrict__ amax, int E, int N) {
    int i = blockIdx.x * blockDim.x + threadIdx.x;
    if (i >= E + N) return;
    float ce = 0.f;
#pragma unroll
    for (int k = 0; k < 9; ++k) ce += We[k] * ae[k];   // (ea @ We) . ae == ea * ce
    int s, d; float attr;
    if (i < E) { s = src[i]; d = dst[i]; attr = ea[i]; }
    else       { s = d = i - E; attr = loop_attr[s]; }  // self loops, fill_value='mean'
    float a = leaky(ss[s] + sd[d] + attr * ce, 0.2f);
    a_buf[i] = a;
    atomicMaxF(&amax[d], a);
}

// ---------------------------------------------------------------- edge pass 2: exp + weighted scatter
__global__ void k_edge_b(const int* __restrict__ src, const int* __restrict__ dst,
                         const float* __restrict__ a_buf, const float* __restrict__ amax,
                         const float* __restrict__ xp, float* __restrict__ denom,
                         float* __restrict__ acc, int E, int N) {
    int i = blockIdx.x * blockDim.x + threadIdx.x;
    if (i >= E + N) return;
    int s, d;
    if (i < E) { s = src[i]; d = dst[i]; } else { s = d = i - E; }
    float aexp = expf(a_buf[i] - amax[d]);
    const float4* xr = (const float4*)(xp + (size_t)s * NS);   // 3x b128 gather
    float4 r0 = xr[0], r1 = xr[1], r2 = xr[2];
    float f[12];
    *(float4*)&f[0] = r0; *(float4*)&f[4] = r1; *(float4*)&f[8] = r2;
    atomicAdd(&denom[d], aexp);
    float* ar = acc + (size_t)d * NS;
#pragma unroll
    for (int k = 0; k < 9; ++k) atomicAdd(&ar[k], aexp * f[k]);
}

// ---------------------------------------------------------------- node finalize, layer 0: leaky only
__global__ void k_fin0(const float* __restrict__ acc, const float* __restrict__ denom,
                       const float* __restrict__ b, float* __restrict__ h1, int N) {
    int t = blockIdx.x * blockDim.x + threadIdx.x;
    if (t >= N) return;
    float dn = denom[t] + 1e-16f;
    const float4* ap = (const float4*)(acc + (size_t)t * NS);
    float4 r0 = ap[0], r1 = ap[1], r2 = ap[2];
    float f[12];
    *(float4*)&f[0] = r0; *(float4*)&f[4] = r1; *(float4*)&f[8] = r2;
#pragma unroll
    for (int k = 0; k < 9; ++k) f[k] = leaky(f[k] / dn + b[k], 0.01f);
    f[9] = f[10] = f[11] = 0.f;
    float4* op = (float4*)(h1 + (size_t)t * NS);
    op[0] = *(float4*)&f[0]; op[1] = *(float4*)&f[4]; op[2] = *(float4*)&f[8];
}

// ---------------------------------------------------------------- node finalize, layer 1: LN + residual
__global__ void k_fin1(const float* __restrict__ acc, const float* __restrict__ denom,
                       const float* __restrict__ b, const float* __restrict__ ng,
                       const float* __restrict__ nb, const float* __restrict__ h1,
                       float* __restrict__ h2, int N) {
    int t = blockIdx.x * blockDim.x + threadIdx.x;
    if (t >= N) return;
    float dn = denom[t] + 1e-16f;
    const float4* ap = (const float4*)(acc + (size_t)t * NS);
    float4 r0 = ap[0], r1 = ap[1], r2 = ap[2];
    float f[12];
    *(float4*)&f[0] = r0; *(float4*)&f[4] = r1; *(float4*)&f[8] = r2;
    float m = 0.f, v = 0.f;
#pragma unroll
    for (int k = 0; k < 9; ++k) { f[k] = f[k] / dn + b[k]; m += f[k]; }
    m *= (1.f / 9.f);
#pragma unroll
    for (int k = 0; k < 9; ++k) { float d = f[k] - m; v += d * d; }
    v *= (1.f / 9.f);
    float inv = rsqrtf(v + EPS_LN);
    const float4* hp = (const float4*)(h1 + (size_t)t * NS);
    float4 q0 = hp[0], q1 = hp[1], q2 = hp[2];
    float r[12];
    *(float4*)&r[0] = q0; *(float4*)&r[4] = q1; *(float4*)&r[8] = q2;
#pragma unroll
    for (int k = 0; k < 9; ++k)
        f[k] = leaky((f[k] - m) * inv * ng[k] + nb[k] + r[k], 0.01f);
    f[9] = f[10] = f[11] = 0.f;
    float4* op = (float4*)(h2 + (size_t)t * NS);
    op[0] = *(float4*)&f[0]; op[1] = *(float4*)&f[4]; op[2] = *(float4*)&f[8];
}

// ---------------------------------------------------------------- edge MLP + attention output
// gather-concat-LN [E,18] -> @et_W (WMMA 16x16x4, K padded 18->20) -> leaky -> @fc_W -> attention
__global__ void k_edge_out(const float* __restrict__ h2, const int* __restrict__ src,
                           const int* __restrict__ dst, const float* __restrict__ ea,
                           const float* __restrict__ g2, const float* __restrict__ b2,
                           const float* __restrict__ etW, const float* __restrict__ etb,
                           const float* __restrict__ fcW, const float* __restrict__ fcb,
                           const float* __restrict__ attW, const float* __restrict__ attb,
                           float* __restrict__ out, int E) {
    __shared__ float Bw[20][16];     // et_W padded
    __shared__ float Af[8][16][20];  // per-wave edge-feature tiles (post-LN)
    __shared__ float Ot[8][16][12];  // per-wave MLP hidden (post-leaky)
    int tid = threadIdx.x;
    int w = tid >> 5, lane = tid & 31, half = lane >> 4, l16 = lane & 15;
    int e = blockIdx.x * 128 + w * 16 + l16;

    for (int i = tid; i < 20 * 16; i += 256) {
        int k = i >> 4, nn = i & 15;
        if (k < 18 && nn < 9) g2l_b32(&etW[k * 9 + nn], &Bw[k][nn]);  // async G->LDS
        else                  Bw[k][nn] = 0.f;
    }
    if (half == 0) {
        float f[18];
        if (e < E) {
            int s = src[e], d = dst[e];
            if (e + 128 < E) __builtin_prefetch(&src[e + 128], 0, 0);  // global_prefetch_b8
            const float4* sp = (const float4*)(h2 + (size_t)s * NS);
            const float4* dp = (const float4*)(h2 + (size_t)d * NS);
            float4 a0 = sp[0], a1 = sp[1], a2 = sp[2];
            float4 c0 = dp[0], c1 = dp[1], c2 = dp[2];
            float fa[12], fb[12];
            *(float4*)&fa[0] = a0; *(float4*)&fa[4] = a1; *(float4*)&fa[8] = a2;
            *(float4*)&fb[0] = c0; *(float4*)&fb[4] = c1; *(float4*)&fb[8] = c2;
#pragma unroll
            for (int k = 0; k < 9; ++k) { f[k] = fa[k]; f[9 + k] = fb[k]; }
            float m = 0.f, v = 0.f;
#pragma unroll
            for (int k = 0; k < 18; ++k) m += f[k];
            m *= (1.f / 18.f);
#pragma unroll
            for (int k = 0; k < 18; ++k) { float t = f[k] - m; v += t * t; }
            v *= (1.f / 18.f);
            float inv = rsqrtf(v + EPS_LN);
#pragma unroll
            for (int k = 0; k < 18; ++k) f[k] = (f[k] - m) * inv * g2[k] + b2[k];
        } else {
#pragma unroll
            for (int k = 0; k < 18; ++k) f[k] = 0.f;
        }
#pragma unroll
        for (int k = 0; k < 20; ++k) Af[w][l16][k] = (k < 18) ? f[k] : 0.f;
    }
    g2l_wait();
    __syncthreads();

    v8f c = {};
#pragma unroll
    for (int ch = 0; ch < 5; ++ch) {
        int kb = ch * 4 + half * 2;
        v2f A; A.x = Af[w][l16][kb]; A.y = Af[w][l16][kb + 1];
        v2f B; B.x = Bw[kb][l16];    B.y = Bw[kb + 1][l16];
        c = __builtin_amdgcn_wmma_f32_16x16x4_f32(false, A, false, B, (short)0, c, false, false);
    }
#pragma unroll
    for (int j = 0; j < 8; ++j) {
        if (l16 < 12) {
            float o = 0.f;
            if (l16 < 9) o = leaky(c[j] + etb[l16], 0.01f);
            Ot[w][j + half * 8][l16] = o;
        }
    }
    __syncthreads();

    if (half == 0 && e < E) {
        float eo = fcb[0];
#pragma unroll
        for (int k = 0; k < 9; ++k) eo += Ot[w][l16][k] * fcW[k];
        float a = ea[e];
        float logit = eo * attW[0] + a * attW[1] + attb[0];
        (void)logit;                 // softmax over length-1 last axis == 1.0
        float att = eo + a;          // (w * comb).sum with w == 1
        out[e] = att + eo;
    }
}

// ================================================================ host side
extern "C" void kernel_launch(void* const* d_in, const int* in_sizes, int n_in,
                              void* d_out, int out_size, void* d_ws, size_t ws_size,
                              hipStream_t stream) {
    // setup_inputs() order
    const float* x     = (const float*)d_in[0];
    const int*   ei    = (const int*)d_in[1];
    const float* ea    = (const float*)d_in[2];
    const float* ln1_g = (const float*)d_in[3];
    const float* ln1_b = (const float*)d_in[4];
    const float* W0  = (const float*)d_in[5];
    const float* as0 = (const float*)d_in[6];
    const float* ad0 = (const float*)d_in[7];
    const float* We0 = (const float*)d_in[8];
    const float* ae0 = (const float*)d_in[9];
    const float* b0  = (const float*)d_in[10];
    const float* W1  = (const float*)d_in[11];
    const float* as1 = (const float*)d_in[12];
    const float* ad1 = (const float*)d_in[13];
    const float* We1 = (const float*)d_in[14];
    const float* ae1 = (const float*)d_in[15];
    const float* b1  = (const float*)d_in[16];
    const float* ng1 = (const float*)d_in[17];
    const float* nb1 = (const float*)d_in[18];
    const float* ln2_g = (const float*)d_in[19];
    const float* ln2_b = (const float*)d_in[20];
    const float* et_W = (const float*)d_in[21];
    const float* et_b = (const float*)d_in[22];
    const float* fc_W = (const float*)d_in[23];
    const float* fc_b = (const float*)d_in[24];
    const float* attW = (const float*)d_in[25];
    const float* attb = (const float*)d_in[26];

    const int N = in_sizes[0] / 9;
    const int E = in_sizes[2];
    const int* src = ei;
    const int* dst = ei + E;
    float* out = (float*)d_out;

    // workspace layout (all buffers 16B-aligned; node features stride NS=12)
    float* ws = (float*)d_ws;
    size_t off = 0;
    size_t Na = (size_t)((N + 3) & ~3);
    float* h0      = ws + off; off += (size_t)N * NS;
    float* xp      = ws + off; off += (size_t)N * NS;
    float* ss      = ws + off; off += Na;
    float* sd      = ws + off; off += Na;
    float* deg     = ws + off; off += Na;
    float* loopat  = ws + off; off += Na;   // loopsum, then loop_attr in place
    float* amax    = ws + off; off += Na;
    float* denom   = ws + off; off += Na;
    float* acc     = ws + off; off += (size_t)N * NS;
    float* h1      = ws + off; off += (size_t)N * NS;
    float* h2      = ws + off; off += (size_t)N * NS;
    float* a_buf   = ws + off; off += (size_t)(E + N);
    (void)ws_size; (void)n_in; (void)out_size;

    const int B = 256;
    const int gN   = (N + B - 1) / B;
    const int gNS  = (N * NS + B - 1) / B;
    const int gE   = (E + B - 1) / B;
    const int gEN  = (E + N + B - 1) / B;
    const int gPRJ = (N + 127) / 128;
    const int gEO  = (E + 127) / 128;

    // ---- prep: LN(x), degree, mean loop attr
    k_fill<<<gN, B, 0, stream>>>(deg, 0.f, N);
    k_fill<<<gN, B, 0, stream>>>(loopat, 0.f, N);
    k_ln9<<<gN, B, 0, stream>>>(x, ln1_g, ln1_b, h0, N);
    k_deg<<<gE, B, 0, stream>>>(dst, ea, deg, loopat, E);
    k_loopattr<<<gN, B, 0, stream>>>(loopat, deg, N);

    // ---- GAT layer 0
    k_fill<<<gN, B, 0, stream>>>(amax, -INFINITY, N);
    k_fill<<<gN, B, 0, stream>>>(denom, 0.f, N);
    k_fill<<<gNS, B, 0, stream>>>(acc, 0.f, N * NS);
    k_proj<<<gPRJ, B, 0, stream>>>(h0, W0, xp, N);
    k_scores<<<gN, B, 0, stream>>>(xp, as0, ad0, ss, sd, N);
    k_edge_a<<<gEN, B, 0, stream>>>(src, dst, ea, loopat, ss, sd, We0, ae0, a_buf, amax, E, N);
    k_edge_b<<<gEN, B, 0, stream>>>(src, dst, a_buf, amax, xp, denom, acc, E, N);
    k_fin0<<<gN, B, 0, stream>>>(acc, denom, b0, h1, N);

    // ---- GAT layer 1 (norm + residual)
    k_fill<<<gN, B, 0, stream>>>(amax, -INFINITY, N);
    k_fill<<<gN, B, 0, stream>>>(denom, 0.f, N);
    k_fill<<<gNS, B, 0, stream>>>(acc, 0.f, N * NS);
    k_proj<<<gPRJ, B, 0, stream>>>(h1, W1, xp, N);
    k_scores<<<gN, B, 0, stream>>>(xp, as1, ad1, ss, sd, N);
    k_edge_a<<<gEN, B, 0, stream>>>(src, dst, ea, loopat, ss, sd, We1, ae1, a_buf, amax, E, N);
    k_edge_b<<<gEN, B, 0, stream>>>(src, dst, a_buf, amax, xp, denom, acc, E, N);
    k_fin1<<<gN, B, 0, stream>>>(acc, denom, b1, ng1, nb1, h1, h2, N);

    // ---- edge path: gather-concat-LN -> MLP (WMMA) -> attention combine
    k_edge_out<<<gEO, B, 0, stream>>>(h2, src, dst, ea, ln2_g, ln2_b, et_W, et_b,
                                      fc_W, fc_b, attW, attb, out, E);
}